// MultiHeadGeneticAttention_13116830122113
// MI455X (gfx1250) — compile-verified
//
#include <hip/hip_runtime.h>

typedef float v2f __attribute__((ext_vector_type(2)));
typedef float v4f __attribute__((ext_vector_type(4)));
typedef float v8f __attribute__((ext_vector_type(8)));

#define TT 2048
#define DD 1024
#define HH 16
#define HD 64
#define WIN 128
#define NTI 128          // TT/16
#define NTJ 9            // j-tiles covering the 129-wide causal window
#define NEGV (-1.0e10f)

#define KC   32          // K chunk staged through LDS
#define LDST 36          // padded LDS row stride (dwords): conflict-free + 16B aligned

// workspace offsets (in floats)
#define OFF_G   0u          // g buffer [H][T][HD] -- reused as ctx later
#define OFF_V   2097152u    // v buffer [H][T][HD]
#define OFF_S   4194304u    // score band [H][NTI][NTJ][16][16]
#define OFF_NRM 8912896u    // norms   [H][T]
#define OFF_DEN 8945664u    // denom   [H][T]
#define OFF_FIT 8978432u    // fitness [H][T]
#define OFF_CTX OFF_G       // context [T][D] aliases g (g dead after scores kernel)

__device__ __forceinline__ v8f wmma4(v2f a, v2f b, v8f c) {
    // D = A(16x4,f32) * B(4x16,f32) + C(16x16,f32)
    return __builtin_amdgcn_wmma_f32_16x16x4_f32(
        false, a, false, b, (short)0, c, false, false);
}

// ---------------------------------------------------------------------------
// Kernel 1: fused g = x*Wg^T + bg, v = x*Wv^T + bv; head-major [H][T][64] out.
// Block = 8 waves = 128 M rows sharing one 32-col weight tile (Wg+Wv) staged
// in double-buffered LDS; next chunk prefetched into registers during compute;
// LDS fragment reads software-pipelined one k-step ahead of the WMMA chain.
// grid (32, 16), block 256.
// ---------------------------------------------------------------------------
__global__ __launch_bounds__(256)
void k_gemm_gv(const float* __restrict__ x, const float* __restrict__ Wg,
               const float* __restrict__ bg, const float* __restrict__ Wv,
               const float* __restrict__ bv, float* __restrict__ ws)
{
    __shared__ __align__(16) float lds[2][64 * LDST]; // rows 0-31: Wg, 32-63: Wv

    const int tid  = threadIdx.x;
    const int lane = tid & 31;
    const int wave = tid >> 5;
    const int lm = lane & 15, lk = lane >> 4;
    const int m0 = blockIdx.y * 128 + wave * 16;   // this wave's 16 M rows
    const int n0 = blockIdx.x * 32;                // block's 32 N cols

    // staging: thread -> (row sr of 32, col group sc of 4) for each weight
    const int sr = tid >> 3;
    const int sc = (tid & 7) * 4;
    const float* gsrc = Wg + (size_t)(n0 + sr) * DD + sc;
    const float* vsrc = Wv + (size_t)(n0 + sr) * DD + sc;

    const float* xr = x + (size_t)(m0 + lm) * DD + 2 * lk;

    v4f stg_g = *(const v4f*)(gsrc);
    v4f stg_v = *(const v4f*)(vsrc);

    v8f cg0 = {}, cg1 = {}, cv0 = {}, cv1 = {};

    int buf = 0;
    for (int kc = 0; kc < DD; kc += KC) {
        // commit staged registers to LDS
        *(v4f*)&lds[buf][sr * LDST + sc]        = stg_g;
        *(v4f*)&lds[buf][(32 + sr) * LDST + sc] = stg_v;
        __syncthreads();
        // prefetch next chunk into registers (overlaps the WMMAs below)
        if (kc + KC < DD) {
            stg_g = *(const v4f*)(gsrc + kc + KC);
            stg_v = *(const v4f*)(vsrc + kc + KC);
        }
        // preload all A fragments of this chunk
        v2f af[8];
#pragma unroll
        for (int s = 0; s < 8; ++s)
            af[s] = *(const v2f*)(xr + kc + 4 * s);

        const float* bgp0 = &lds[buf][(lm)      * LDST + 2 * lk];
        const float* bgp1 = &lds[buf][(16 + lm) * LDST + 2 * lk];
        const float* bvp0 = &lds[buf][(32 + lm) * LDST + 2 * lk];
        const float* bvp1 = &lds[buf][(48 + lm) * LDST + 2 * lk];

        // 1-step software pipeline of the LDS fragment reads
        v2f q0 = *(const v2f*)(bgp0);
        v2f q1 = *(const v2f*)(bgp1);
        v2f q2 = *(const v2f*)(bvp0);
        v2f q3 = *(const v2f*)(bvp1);
#pragma unroll
        for (int s = 0; s < 8; ++s) {
            v2f t0, t1, t2, t3;
            if (s < 7) {
                t0 = *(const v2f*)(bgp0 + 4 * (s + 1));
                t1 = *(const v2f*)(bgp1 + 4 * (s + 1));
                t2 = *(const v2f*)(bvp0 + 4 * (s + 1));
                t3 = *(const v2f*)(bvp1 + 4 * (s + 1));
            }
            cg0 = wmma4(af[s], q0, cg0);
            cg1 = wmma4(af[s], q1, cg1);
            cv0 = wmma4(af[s], q2, cv0);
            cv1 = wmma4(af[s], q3, cv1);
            if (s < 7) { q0 = t0; q1 = t1; q2 = t2; q3 = t3; }
        }
        __syncthreads();
        buf ^= 1;
    }

    const int c0 = n0 + lm, c1 = n0 + 16 + lm;
    const float bgv0 = bg[c0], bgv1 = bg[c1];
    const float bvv0 = bv[c0], bvv1 = bv[c1];
    float* gbuf = ws + OFF_G;
    float* vbuf = ws + OFF_V;
#pragma unroll
    for (int r = 0; r < 8; ++r) {
        const int t = m0 + r + 8 * lk;
        gbuf[(size_t)(c0 >> 6) * (TT * HD) + t * HD + (c0 & 63)] = cg0[r] + bgv0;
        gbuf[(size_t)(c1 >> 6) * (TT * HD) + t * HD + (c1 & 63)] = cg1[r] + bgv1;
        vbuf[(size_t)(c0 >> 6) * (TT * HD) + t * HD + (c0 & 63)] = cv0[r] + bvv0;
        vbuf[(size_t)(c1 >> 6) * (TT * HD) + t * HD + (c1 & 63)] = cv1[r] + bvv1;
    }
}

// ---------------------------------------------------------------------------
// Kernel 2: per-row L2 norms of g. One wave per (h,t) row of 64. grid 4096.
// ---------------------------------------------------------------------------
__global__ __launch_bounds__(256)
void k_norms(const float* __restrict__ gbuf, float* __restrict__ nrm)
{
    const int lane = threadIdx.x & 31;
    const int wave = threadIdx.x >> 5;
    const int row  = blockIdx.x * 8 + wave;      // [0, H*T)
    const float* gr = gbuf + (size_t)row * HD;
    float a = gr[lane], b = gr[lane + 32];
    float s = a * a + b * b;
#pragma unroll
    for (int off = 16; off > 0; off >>= 1) s += __shfl_xor(s, off, 32);
    if (lane == 0) nrm[row] = sqrtf(s);
}

// ---------------------------------------------------------------------------
// Kernel 3: banded cosine scores via WMMA. One wave per (h, i-tile).
// A fragments (16) hoisted across all 9 j-tiles; per tile all 16 B fragments
// are issued before the WMMA chain so waits pipeline (loadcnt 15..0).
// grid 256, block 256. Masked entries stored as 0.
// ---------------------------------------------------------------------------
__global__ __launch_bounds__(256)
void k_scores(const float* __restrict__ gbuf, const float* __restrict__ nrm,
              float* __restrict__ scores)
{
    const int lane = threadIdx.x & 31;
    const int wave = threadIdx.x >> 5;
    const int lm = lane & 15, lk = lane >> 4;
    const int gid = blockIdx.x * 8 + wave;       // [0, H*NTI)
    const int h = gid >> 7, ti = gid & 127;
    const int i0 = ti * 16;

    const float* ghead = gbuf + (size_t)h * (TT * HD);
    const float* arow  = ghead + (size_t)(i0 + lm) * HD + 2 * lk;

    v2f areg[16];
#pragma unroll
    for (int s = 0; s < 16; ++s) areg[s] = *(const v2f*)(arow + 4 * s);

    float ni[8];
#pragma unroll
    for (int r = 0; r < 8; ++r) ni[r] = nrm[h * TT + i0 + r + 8 * lk];

    float* sbase = scores + (size_t)gid * (NTJ * 256);

    for (int tj = 0; tj < NTJ; ++tj) {
        float* stile = sbase + tj * 256;
        const int tja = ti - 8 + tj;
        if (tja < 0) {                            // fully out of window
#pragma unroll
            for (int r = 0; r < 8; ++r) stile[(r + 8 * lk) * 16 + lm] = 0.0f;
            continue;
        }
        const int j0 = tja * 16;
        const float* brow = ghead + (size_t)(j0 + lm) * HD + 2 * lk;
        v2f breg[16];
#pragma unroll
        for (int s = 0; s < 16; ++s) breg[s] = *(const v2f*)(brow + 4 * s);
        v8f c = {};
#pragma unroll
        for (int s = 0; s < 16; ++s) c = wmma4(areg[s], breg[s], c);

        const float nj = nrm[h * TT + j0 + lm];
#pragma unroll
        for (int r = 0; r < 8; ++r) {
            const int m  = r + 8 * lk;
            const int iA = i0 + m, jA = j0 + lm;
            float cosv = c[r] / (ni[r] * nj + 1e-8f);
            const bool in = (jA <= iA) && (iA - jA <= WIN);
            stile[m * 16 + lm] = in ? cosv : 0.0f;
        }
    }
}

// ---------------------------------------------------------------------------
// Kernel 4: per-head fitness. denom_i = 1 + S_i/(2T);
// fitness_i = 1/(denom_i * sum_i 1/denom_i). One block per head (grid 16).
// ---------------------------------------------------------------------------
__global__ __launch_bounds__(256)
void k_stats(const float* __restrict__ scores, float* __restrict__ den,
             float* __restrict__ fit)
{
    __shared__ float red[256];
    const int h = blockIdx.x, tid = threadIdx.x;
    float inv = 0.0f;
    float dloc[8];
#pragma unroll
    for (int r = 0; r < 8; ++r) {
        const int i = tid + 256 * r;              // row within head, [0, T)
        const int ti = i >> 4, m = i & 15;
        const float* row = scores + (size_t)(h * NTI + ti) * (NTJ * 256) + m * 16;
        float s = 0.0f;
        for (int tj = 0; tj < NTJ; ++tj)
#pragma unroll
            for (int n = 0; n < 16; ++n) s += row[tj * 256 + n];
        const float dd = 1.0f + s * (1.0f / (2.0f * TT));
        dloc[r] = dd;
        inv += 1.0f / dd;
        den[h * TT + i] = dd;
    }
    red[tid] = inv;
    __syncthreads();
    for (int off = 128; off > 0; off >>= 1) {
        if (tid < off) red[tid] += red[tid + off];
        __syncthreads();
    }
    const float suminv = red[0];
#pragma unroll
    for (int r = 0; r < 8; ++r) {
        const int i = tid + 256 * r;
        fit[h * TT + i] = 1.0f / (dloc[r] * suminv);
    }
}

// ---------------------------------------------------------------------------
// Kernel 5: in-place band softmax of ((cos+1)/2)*fitness; masked -> 0.
// One thread per (h,i) row. grid 128, block 256.
// ---------------------------------------------------------------------------
__global__ __launch_bounds__(256)
void k_softmax(float* __restrict__ scores, const float* __restrict__ fit)
{
    const int idx = blockIdx.x * 256 + threadIdx.x;   // [0, H*T)
    const int h = idx >> 11, i = idx & 2047;
    const int ti = i >> 4, m = i & 15;
    const float f = fit[idx];
    float* row = scores + (size_t)(h * NTI + ti) * (NTJ * 256) + m * 16;

    float mx = -3.4e38f;
    for (int tj = 0; tj < NTJ; ++tj) {
        const int j0 = (ti - 8 + tj) * 16;
#pragma unroll
        for (int n = 0; n < 16; ++n) {
            const int j = j0 + n;
            const bool in = (j >= 0) && (j <= i) && (i - j <= WIN);
            const float val = in ? 0.5f * (row[tj * 256 + n] + 1.0f) * f : NEGV;
            mx = fmaxf(mx, val);
        }
    }
    float se = 0.0f;
    for (int tj = 0; tj < NTJ; ++tj) {
        const int j0 = (ti - 8 + tj) * 16;
#pragma unroll
        for (int n = 0; n < 16; ++n) {
            const int j = j0 + n;
            const bool in = (j >= 0) && (j <= i) && (i - j <= WIN);
            if (in) se += __expf(0.5f * (row[tj * 256 + n] + 1.0f) * f - mx);
        }
    }
    const float rs = 1.0f / se;
    for (int tj = 0; tj < NTJ; ++tj) {
        const int j0 = (ti - 8 + tj) * 16;
#pragma unroll
        for (int n = 0; n < 16; ++n) {
            const int j = j0 + n;
            const bool in = (j >= 0) && (j <= i) && (i - j <= WIN);
            row[tj * 256 + n] =
                in ? __expf(0.5f * (row[tj * 256 + n] + 1.0f) * f - mx) * rs : 0.0f;
        }
    }
}

// ---------------------------------------------------------------------------
// Kernel 6: ctx = P @ V via WMMA. One wave per (h, i-tile): 16x64 output over
// 9 band tiles. All A (4) and B (16) fragments of a tile are loaded before the
// 16-WMMA chain. grid 256, block 256. Writes context in [T][D].
// ---------------------------------------------------------------------------
__global__ __launch_bounds__(256)
void k_av(const float* __restrict__ scores, const float* __restrict__ vbuf,
          float* __restrict__ ctx)
{
    const int lane = threadIdx.x & 31;
    const int wave = threadIdx.x >> 5;
    const int lm = lane & 15, lk = lane >> 4;
    const int gid = blockIdx.x * 8 + wave;
    const int h = gid >> 7, ti = gid & 127;
    const int i0 = ti * 16;

    const float* vh    = vbuf + (size_t)h * (TT * HD);
    const float* sbase = scores + (size_t)gid * (NTJ * 256);

    v8f a0 = {}, a1 = {}, a2 = {}, a3 = {};
    for (int tj = 0; tj < NTJ; ++tj) {
        const int tja = ti - 8 + tj;
        if (tja < 0) continue;                    // P is all-zero there
        const int j0 = tja * 16;
        const float* ptile = sbase + tj * 256;

        v2f ap[4];
#pragma unroll
        for (int s = 0; s < 4; ++s)
            ap[s] = *(const v2f*)(ptile + lm * 16 + 4 * s + 2 * lk);

        v2f bf[16];
#pragma unroll
        for (int s = 0; s < 4; ++s) {
            const float* vr = vh + (size_t)(j0 + 4 * s + 2 * lk) * HD + lm;
#pragma unroll
            for (int nt = 0; nt < 4; ++nt) {
                v2f b; b.x = vr[nt * 16]; b.y = vr[HD + nt * 16];
                bf[s * 4 + nt] = b;
            }
        }
#pragma unroll
        for (int s = 0; s < 4; ++s) {
            a0 = wmma4(ap[s], bf[s * 4 + 0], a0);
            a1 = wmma4(ap[s], bf[s * 4 + 1], a1);
            a2 = wmma4(ap[s], bf[s * 4 + 2], a2);
            a3 = wmma4(ap[s], bf[s * 4 + 3], a3);
        }
    }
#pragma unroll
    for (int r = 0; r < 8; ++r) {
        const int t = i0 + r + 8 * lk;
        float* o = ctx + (size_t)t * DD + h * HD;
        o[lm]      = a0[r];
        o[16 + lm] = a1[r];
        o[32 + lm] = a2[r];
        o[48 + lm] = a3[r];
    }
}

// ---------------------------------------------------------------------------
// Kernel 7: out = ctx @ Wo^T + bo. Same LDS-tiled structure as kernel 1:
// block = 128 M rows x 64 N cols, Wo chunk staged in double-buffered LDS;
// LDS fragment reads software-pipelined one k-step ahead.
// grid (16, 16), block 256.
// ---------------------------------------------------------------------------
__global__ __launch_bounds__(256)
void k_gemm_out(const float* __restrict__ ctx, const float* __restrict__ Wo,
                const float* __restrict__ bo, float* __restrict__ out)
{
    __shared__ __align__(16) float lds[2][64 * LDST]; // 64 Wo rows x KC cols

    const int tid  = threadIdx.x;
    const int lane = tid & 31;
    const int wave = tid >> 5;
    const int lm = lane & 15, lk = lane >> 4;
    const int m0 = blockIdx.y * 128 + wave * 16;
    const int n0 = blockIdx.x * 64;

    // staging: thread -> (row sr of 64, col group sc of 8 -> two v4f)
    const int sr = tid >> 2;
    const int sc = (tid & 3) * 8;
    const float* wsrc = Wo + (size_t)(n0 + sr) * DD + sc;

    const float* ar = ctx + (size_t)(m0 + lm) * DD + 2 * lk;

    v4f stg0 = *(const v4f*)(wsrc);
    v4f stg1 = *(const v4f*)(wsrc + 4);

    v8f c0 = {}, c1 = {}, c2 = {}, c3 = {};

    int buf = 0;
    for (int kc = 0; kc < DD; kc += KC) {
        *(v4f*)&lds[buf][sr * LDST + sc]     = stg0;
        *(v4f*)&lds[buf][sr * LDST + sc + 4] = stg1;
        __syncthreads();
        if (kc + KC < DD) {
            stg0 = *(const v4f*)(wsrc + kc + KC);
            stg1 = *(const v4f*)(wsrc + kc + KC + 4);
        }
        v2f af[8];
#pragma unroll
        for (int s = 0; s < 8; ++s)
            af[s] = *(const v2f*)(ar + kc + 4 * s);

        const float* b0 = &lds[buf][(lm)      * LDST + 2 * lk];
        const float* b1 = &lds[buf][(16 + lm) * LDST + 2 * lk];
        const float* b2 = &lds[buf][(32 + lm) * LDST + 2 * lk];
        const float* b3 = &lds[buf][(48 + lm) * LDST + 2 * lk];

        v2f q0 = *(const v2f*)(b0);
        v2f q1 = *(const v2f*)(b1);
        v2f q2 = *(const v2f*)(b2);
        v2f q3 = *(const v2f*)(b3);
#pragma unroll
        for (int s = 0; s < 8; ++s) {
            v2f t0, t1, t2, t3;
            if (s < 7) {
                t0 = *(const v2f*)(b0 + 4 * (s + 1));
                t1 = *(const v2f*)(b1 + 4 * (s + 1));
                t2 = *(const v2f*)(b2 + 4 * (s + 1));
                t3 = *(const v2f*)(b3 + 4 * (s + 1));
            }
            c0 = wmma4(af[s], q0, c0);
            c1 = wmma4(af[s], q1, c1);
            c2 = wmma4(af[s], q2, c2);
            c3 = wmma4(af[s], q3, c3);
            if (s < 7) { q0 = t0; q1 = t1; q2 = t2; q3 = t3; }
        }
        __syncthreads();
        buf ^= 1;
    }

    const float bb0 = bo[n0 + lm],      bb1 = bo[n0 + 16 + lm];
    const float bb2 = bo[n0 + 32 + lm], bb3 = bo[n0 + 48 + lm];
#pragma unroll
    for (int r = 0; r < 8; ++r) {
        const int t = m0 + r + 8 * lk;
        float* o = out + (size_t)t * DD + n0;
        o[lm]      = c0[r] + bb0;
        o[16 + lm] = c1[r] + bb1;
        o[32 + lm] = c2[r] + bb2;
        o[48 + lm] = c3[r] + bb3;
    }
}

// ---------------------------------------------------------------------------
extern "C" void kernel_launch(void* const* d_in, const int* in_sizes, int n_in,
                              void* d_out, int out_size, void* d_ws, size_t ws_size,
                              hipStream_t stream)
{
    (void)in_sizes; (void)n_in; (void)out_size; (void)ws_size;
    const float* x  = (const float*)d_in[0];
    const float* Wg = (const float*)d_in[1];
    const float* bg = (const float*)d_in[2];
    const float* Wv = (const float*)d_in[3];
    const float* bv = (const float*)d_in[4];
    const float* Wo = (const float*)d_in[5];
    const float* bo = (const float*)d_in[6];
    float* out = (float*)d_out;
    float* ws  = (float*)d_ws;

    k_gemm_gv <<<dim3(32, 16), 256, 0, stream>>>(x, Wg, bg, Wv, bv, ws);
    k_norms   <<<dim3(4096),   256, 0, stream>>>(ws + OFF_G, ws + OFF_NRM);
    k_scores  <<<dim3(256),    256, 0, stream>>>(ws + OFF_G, ws + OFF_NRM, ws + OFF_S);
    k_stats   <<<dim3(16),     256, 0, stream>>>(ws + OFF_S, ws + OFF_DEN, ws + OFF_FIT);
    k_softmax <<<dim3(128),    256, 0, stream>>>(ws + OFF_S, ws + OFF_FIT);
    k_av      <<<dim3(256),    256, 0, stream>>>(ws + OFF_S, ws + OFF_V, ws + OFF_CTX);
    k_gemm_out<<<dim3(16, 16), 256, 0, stream>>>(ws + OFF_CTX, Wo, bo, out);
}